// ElasticTransform_11115375362426
// MI455X (gfx1250) — compile-verified
//
#include <hip/hip_runtime.h>
#include <hip/hip_bf16.h>

typedef __attribute__((ext_vector_type(2))) float v2f;
typedef __attribute__((ext_vector_type(8))) float v8f;

namespace et {
constexpr int kH = 512, kW = 512, kHW = kH * kW;
constexpr int kB = 8, kC = 31;
constexpr int kKS = 63, kPad = 31;
constexpr float kSigma = 32.0f;
constexpr int kGridOff = kB * kC * kHW;  // 65,011,712 floats: start of grid output
}
using namespace et;

// ---- Sentinel-padded tap table in LDS ----
// k1d[0] = 0, k1d[e] = exp(-(e-1)^2/2048) for e in 1..63, k1d[64] = 0.
__device__ __forceinline__ float tap_val(int e) {
  const float t = (float)(e - 1);
  return (e == 0) ? 0.0f : expf(-t * t * (1.0f / 2048.0f));
}
__device__ __forceinline__ void fill_k1d(float* k1d, int lane) {
  k1d[lane] = tap_val(lane);            // 0..31
  k1d[lane + 32] = tap_val(lane + 32);  // 32..63
  if (lane == 0) k1d[64] = 0.0f;
  __syncthreads();
}
// Clamp tap t into sentinel table index [0,64] (valid taps t=0..62 -> idx 1..63).
__device__ __forceinline__ int tap_clamp(int t) {
  int idx = t + 1;
  idx = idx < 0 ? 0 : idx;
  return idx > 64 ? 64 : idx;
}
// Weight of input line i contributing to output line o (reflect-folded band).
// Branch-free: reflection guards select the zero-sentinel index.
__device__ __forceinline__ float band_w(const float* k1d, int o, int i, int n) {
  const int i0 = tap_clamp(i - o + kPad);                       // direct
  const int i1 = (i > 0) ? tap_clamp(kPad - i - o) : 0;         // low reflection
  const int i2 = (i < n - 1) ? tap_clamp(2 * (n - 1) - i - o + kPad) : 0;  // high
  return k1d[i0] + k1d[i1] + k1d[i2];
}

// ---- Pass 1: vertical blur as band GEMM, D(16x16) += A(weights 16x4) x B(data 4x16)
__global__ __launch_bounds__(32) void blur_v_kernel(
    const float* __restrict__ dxn, const float* __restrict__ dyn,
    float* __restrict__ out) {
  __shared__ float k1d[72];
  const int lane = threadIdx.x;
  fill_k1d(k1d, lane);

  const int x0 = blockIdx.x << 4;
  const int y0 = blockIdx.y << 4;
  const int b = blockIdx.z >> 1;
  const int map = blockIdx.z & 1;
  const float* in = (map ? dyn : dxn) + (size_t)b * kHW;      // uniform base
  float* outp = out + (size_t)(b * kC + 2 + map) * kHW;       // stash in ch 2/3

  const int m = lane & 15;             // M (and N) index for this lane
  const int khalf = (lane >> 4) << 1;  // K sub-index: 0 (lanes 0-15) / 2 (16-31)

  v8f acc = {0.f, 0.f, 0.f, 0.f, 0.f, 0.f, 0.f, 0.f};
  const int kt0 = (y0 > kPad) ? ((y0 - kPad) >> 4) : 0;
  int kt1 = (y0 + 15 + kPad) >> 4;
  if (kt1 > kH / 16 - 1) kt1 = kH / 16 - 1;
  for (int kt = kt0; kt <= kt1; ++kt) {
#pragma unroll
    for (int kb = 0; kb < 16; kb += 4) {
      const int r0 = (kt << 4) + kb + khalf;  // absolute input rows r0, r0+1
      v2f a, bm;
      a.x = band_w(k1d, y0 + m, r0, kH);
      a.y = band_w(k1d, y0 + m, r0 + 1, kH);
      bm.x = in[r0 * kW + x0 + m];            // 32-bit offsets, uniform base
      bm.y = in[(r0 + 1) * kW + x0 + m];
      acc = __builtin_amdgcn_wmma_f32_16x16x4_f32(false, a, false, bm,
                                                  (short)0, acc, false, false);
    }
  }
  const int rbase = y0 + ((lane >> 4) << 3);  // D: vgpr r -> M=r (lo) / r+8 (hi)
#pragma unroll
  for (int r = 0; r < 8; ++r)
    outp[(rbase + r) * kW + x0 + m] = acc[r];
}

// ---- Pass 2: horizontal blur, D(16x16) += A(data 16x4) x B(weights 4x16)
__global__ __launch_bounds__(32) void blur_h_kernel(float* __restrict__ out) {
  __shared__ float k1d[72];
  const int lane = threadIdx.x;
  fill_k1d(k1d, lane);

  const int x0 = blockIdx.x << 4;
  const int y0 = blockIdx.y << 4;
  const int b = blockIdx.z >> 1;
  const int map = blockIdx.z & 1;
  const float* in = out + (size_t)(b * kC + 2 + map) * kHW;  // vert result
  float* outp = out + (size_t)(b * kC + 0 + map) * kHW;      // final -> ch 0/1

  const int m = lane & 15;
  const int khalf = (lane >> 4) << 1;

  v8f acc = {0.f, 0.f, 0.f, 0.f, 0.f, 0.f, 0.f, 0.f};
  const int kt0 = (x0 > kPad) ? ((x0 - kPad) >> 4) : 0;
  int kt1 = (x0 + 15 + kPad) >> 4;
  if (kt1 > kW / 16 - 1) kt1 = kW / 16 - 1;
  for (int kt = kt0; kt <= kt1; ++kt) {
#pragma unroll
    for (int kb = 0; kb < 16; kb += 4) {
      const int c0 = (kt << 4) + kb + khalf;  // absolute input cols c0, c0+1
      v2f a = *(const v2f*)&in[(y0 + m) * kW + c0];  // contiguous pair (8B aligned)
      v2f bm;
      bm.x = band_w(k1d, x0 + m, c0, kW);
      bm.y = band_w(k1d, x0 + m, c0 + 1, kW);
      acc = __builtin_amdgcn_wmma_f32_16x16x4_f32(false, a, false, bm,
                                                  (short)0, acc, false, false);
    }
  }
  const int rbase = y0 + ((lane >> 4) << 3);
#pragma unroll
  for (int r = 0; r < 8; ++r)
    outp[(rbase + r) * kW + x0 + m] = acc[r];
}

// ---- Order-preserving float->uint key for atomic max over signed floats ----
__device__ __forceinline__ unsigned fkey(float f) {
  unsigned u = __float_as_uint(f);
  return (u & 0x80000000u) ? ~u : (u | 0x80000000u);
}
__device__ __forceinline__ float fkey_inv(unsigned u) {
  return __uint_as_float((u & 0x80000000u) ? (u & 0x7FFFFFFFu) : ~u);
}

__global__ void init_kernel(unsigned* slots) { slots[0] = 0u; slots[1] = 0u; }

__global__ __launch_bounds__(256) void reduce_max_kernel(
    const float* __restrict__ out, unsigned* __restrict__ slots) {
  const int map = blockIdx.y;
  unsigned best = 0u;
  const int n = kB * kHW;  // 2^21
  for (int i = blockIdx.x * blockDim.x + threadIdx.x; i < n;
       i += gridDim.x * blockDim.x) {
    const int b = i >> 18;  // kHW == 2^18
    const int p = i & (kHW - 1);
    const float v = out[(size_t)(b * kC + map) * kHW + p];
    const unsigned k = fkey(v);
    best = best > k ? best : k;
  }
  __shared__ unsigned s[256];
  s[threadIdx.x] = best;
  __syncthreads();
  for (int off = 128; off > 0; off >>= 1) {
    if ((int)threadIdx.x < off)
      s[threadIdx.x] = s[threadIdx.x] > s[threadIdx.x + off]
                           ? s[threadIdx.x] : s[threadIdx.x + off];
    __syncthreads();
  }
  if (threadIdx.x == 0) atomicMax(&slots[map], s[0]);
}

// ---- Pass 3: grid build + bilinear border gather over 31 channels ----
__global__ __launch_bounds__(256) void warp_kernel(
    const float* __restrict__ xin, float* out,
    const unsigned* __restrict__ slots) {
  const int i = blockIdx.x * blockDim.x + threadIdx.x;  // 0 .. 8*HW-1
  const int b = i >> 18;
  const int p = i & (kHW - 1);
  const int y = p >> 9;
  const int xc = p & (kW - 1);

  const float maxdx = fkey_inv(slots[0]);
  const float maxdy = fkey_inv(slots[1]);
  // Read own-pixel blurred displacements (aliased in channels 0/1) BEFORE writes.
  const float dxv = out[(size_t)(b * kC + 0) * kHW + p] * kSigma / maxdx;
  const float dyv = out[(size_t)(b * kC + 1) * kHW + p] * kSigma / maxdy;

  const float gx = (float)xc + dxv;
  const float gy = (float)y + dyv;
  const float gridx = 2.0f * gx / (float)(kW - 1) - 1.0f;
  const float gridy = 2.0f * gy / (float)(kH - 1) - 1.0f;
  __builtin_nontemporal_store(gridx, &out[(size_t)kGridOff + (size_t)i * 2 + 0]);
  __builtin_nontemporal_store(gridy, &out[(size_t)kGridOff + (size_t)i * 2 + 1]);

  float ix = (gridx + 1.0f) * 0.5f * (float)(kW - 1);
  float iy = (gridy + 1.0f) * 0.5f * (float)(kH - 1);
  ix = fminf(fmaxf(ix, 0.0f), (float)(kW - 1));
  iy = fminf(fmaxf(iy, 0.0f), (float)(kH - 1));
  const float fx0 = floorf(ix), fy0 = floorf(iy);
  const float wx = ix - fx0, wy = iy - fy0;
  const int x0i = (int)fx0, y0i = (int)fy0;
  const int x1i = min(x0i + 1, kW - 1), y1i = min(y0i + 1, kH - 1);
  const float w00 = (1.0f - wx) * (1.0f - wy);
  const float w01 = wx * (1.0f - wy);
  const float w10 = (1.0f - wx) * wy;
  const float w11 = wx * wy;

  const int o00 = y0i * kW + x0i, o01 = y0i * kW + x1i;
  const int o10 = y1i * kW + x0i, o11 = y1i * kW + x1i;
  const float* img = xin + (size_t)b * kC * kHW;
#pragma unroll 1
  for (int c = 0; c < kC; ++c) {
    const float* ch = img + (size_t)c * kHW;
    if (c + 1 < kC) __builtin_prefetch(ch + kHW + o00, 0, 0);
    const float v = ch[o00] * w00 + ch[o01] * w01 + ch[o10] * w10 + ch[o11] * w11;
    __builtin_nontemporal_store(v, &out[(size_t)(b * kC + c) * kHW + p]);
  }
}

extern "C" void kernel_launch(void* const* d_in, const int* in_sizes, int n_in,
                              void* d_out, int out_size, void* d_ws, size_t ws_size,
                              hipStream_t stream) {
  const float* xin = (const float*)d_in[0];
  const float* dxn = (const float*)d_in[1];
  const float* dyn = (const float*)d_in[2];
  float* out = (float*)d_out;
  unsigned* slots = (unsigned*)d_ws;  // 8 bytes

  init_kernel<<<1, 1, 0, stream>>>(slots);
  dim3 bgrid(kW / 16, kH / 16, kB * 2);
  blur_v_kernel<<<bgrid, 32, 0, stream>>>(dxn, dyn, out);
  blur_h_kernel<<<bgrid, 32, 0, stream>>>(out);
  reduce_max_kernel<<<dim3(1024, 2), 256, 0, stream>>>(out, slots);
  warp_kernel<<<(kB * kHW) / 256, 256, 0, stream>>>(xin, out, slots);
}